// LSM_48558900249084
// MI455X (gfx1250) — compile-verified
//
#include <hip/hip_runtime.h>

typedef __attribute__((ext_vector_type(2))) float v2f;
typedef __attribute__((ext_vector_type(8))) float v8f;

#define S_TOT  12288
#define D_DIM  64
#define N_EDGE 1000000
#define EPSF   1e-6f
#define BLK    128
#define LDST   68            // padded LDS row stride (dwords) to stagger banks
#define GRID1  (S_TOT / BLK) // 96
#define NPB    (GRID1 * GRID1)
#define NEB    4096

// ---------------------------------------------------------------------------
// Kernel 1: gather per-sample beta and row norms. One wave per sample row.
// lane d loads dims {d, d+32}; shuffle-reduce across the wave (wave32).
// ---------------------------------------------------------------------------
__global__ __launch_bounds__(256)
void LSM_norms_kernel(const float* __restrict__ beta,
                      const float* __restrict__ Z,
                      const int*   __restrict__ sidx,
                      float* __restrict__ na, float* __restrict__ nz,
                      float* __restrict__ bs)
{
    const int wid  = (blockIdx.x * blockDim.x + threadIdx.x) >> 5; // sample id
    const int lane = threadIdx.x & 31;
    if (wid >= S_TOT) return;
    const int idx = sidx[wid];
    const float* row = Z + (size_t)idx * D_DIM;
    const float z1 = row[lane], z2 = row[lane + 32];
    const float a1 = z1 + EPSF, a2 = z2 + EPSF;
    float pna = a1 * a1 + a2 * a2;
    float pnz = z1 * z1 + z2 * z2;
    #pragma unroll
    for (int off = 16; off; off >>= 1) {
        pna += __shfl_xor(pna, off, 32);
        pnz += __shfl_xor(pnz, off, 32);
    }
    if (lane == 0) { na[wid] = pna; nz[wid] = pnz; bs[wid] = beta[idx]; }
}

// ---------------------------------------------------------------------------
// Kernel 2: pairwise block term. 128x128 block per 8-wave workgroup.
// Gram tile via V_WMMA_F32_16X16X4_F32 (16 chained WMMAs, K=64).
// ---------------------------------------------------------------------------
__global__ __launch_bounds__(256)
void LSM_pair_kernel(const float* __restrict__ Z,
                     const int*   __restrict__ sidx,
                     const float* __restrict__ na, const float* __restrict__ nz,
                     const float* __restrict__ bs,
                     float* __restrict__ partial)
{
    __shared__ __align__(16) float Zi[BLK * LDST];  // a = z + eps rows
    __shared__ __align__(16) float Zj[BLK * LDST];  // raw z rows
    __shared__ float naS[BLK], nzS[BLK], biS[BLK], bjS[BLK];
    __shared__ float red[256];

    const int tid = threadIdx.x;
    const int jb  = blockIdx.x, ib = blockIdx.y;
    const int i0  = ib * BLK,   j0 = jb * BLK;

    // Cooperative gather: 128 rows x 64 f32 per block side, float4 vectorized.
    #pragma unroll
    for (int it = 0; it < 8; ++it) {
        const int l   = it * 256 + tid;   // 0..2047
        const int row = l >> 4;           // 0..127
        const int c4  = l & 15;           // float4 column
        const int gi  = sidx[i0 + row];
        const float4 v = ((const float4*)(Z + (size_t)gi * D_DIM))[c4];
        float* di = &Zi[row * LDST + c4 * 4];
        di[0] = v.x + EPSF; di[1] = v.y + EPSF; di[2] = v.z + EPSF; di[3] = v.w + EPSF;
        const int gj  = sidx[j0 + row];
        const float4 w = ((const float4*)(Z + (size_t)gj * D_DIM))[c4];
        float* dj = &Zj[row * LDST + c4 * 4];
        dj[0] = w.x; dj[1] = w.y; dj[2] = w.z; dj[3] = w.w;
    }
    if (tid < BLK) { naS[tid] = na[i0 + tid]; biS[tid] = bs[i0 + tid]; }
    else           { const int t = tid - BLK; nzS[t] = nz[j0 + t]; bjS[t] = bs[j0 + t]; }
    __syncthreads();

    const int wave = tid >> 5;      // ti strip (0..7)
    const int lane = tid & 31;
    const int m16  = lane & 15;
    const int h    = lane >> 4;     // half-wave selects K pair / row-half

    // Preload all 16 A fragments for this wave's 16-row strip (32 VGPRs).
    v2f af[16];
    {
        const float* ap = &Zi[(wave * 16 + m16) * LDST + 2 * h];
        #pragma unroll
        for (int ks = 0; ks < 16; ++ks) af[ks] = *(const v2f*)(ap + 4 * ks);
    }

    float acc = 0.0f;
    for (int tj = 0; tj < 8; ++tj) {
        v8f c = {};
        const float* bp = &Zj[(tj * 16 + m16) * LDST + 2 * h];
        #pragma unroll
        for (int ks = 0; ks < 16; ++ks) {
            const v2f bf = *(const v2f*)(bp + 4 * ks);
            c = __builtin_amdgcn_wmma_f32_16x16x4_f32(
                    false, af[ks], false, bf, (short)0, c, false, false);
        }
        // C/D layout: VGPR r -> M = r + 8*(lane>=16), N = lane%16
        #pragma unroll
        for (int r = 0; r < 8; ++r) {
            const int rowb = wave * 16 + r + 8 * h;
            const int colb = tj * 16 + m16;
            const float sq   = naS[rowb] + nzS[colb] - 2.0f * c[r];
            const float dist = sqrtf(fmaxf(sq, 0.0f));
            float v = __expf(biS[rowb] + bjS[colb] - dist);
            if (i0 + rowb == j0 + colb) v = 0.0f;  // positional diagonal
            acc += v;
        }
    }

    red[tid] = acc;
    __syncthreads();
    for (int s = 128; s; s >>= 1) { if (tid < s) red[tid] += red[tid + s]; __syncthreads(); }
    if (tid == 0) partial[blockIdx.y * gridDim.x + blockIdx.x] = red[0];
}

// ---------------------------------------------------------------------------
// Kernel 3: edge term. One wave per edge; grid-stride; L2-resident rows.
// ---------------------------------------------------------------------------
__global__ __launch_bounds__(256)
void LSM_edge_kernel(const float* __restrict__ beta,
                     const float* __restrict__ Z,
                     const int* __restrict__ ei, const int* __restrict__ ej,
                     float* __restrict__ partial)
{
    __shared__ float red[256];
    const int tid  = threadIdx.x;
    const int lane = tid & 31;
    const int gw   = (int)((blockIdx.x * blockDim.x + tid) >> 5);
    const int nw   = (int)((gridDim.x * blockDim.x) >> 5);
    float acc = 0.0f;
    for (int e = gw; e < N_EDGE; e += nw) {
        const int i = ei[e], j = ej[e];
        const float* zi = Z + (size_t)i * D_DIM;
        const float* zj = Z + (size_t)j * D_DIM;
        const float d1 = zi[lane]      - zj[lane]      + EPSF;
        const float d2 = zi[lane + 32] - zj[lane + 32] + EPSF;
        float ss = d1 * d1 + d2 * d2;
        #pragma unroll
        for (int off = 16; off; off >>= 1) ss += __shfl_xor(ss, off, 32);
        if (lane == 0) acc += beta[i] + beta[j] - sqrtf(ss);
    }
    red[tid] = acc;
    __syncthreads();
    for (int s = 128; s; s >>= 1) { if (tid < s) red[tid] += red[tid + s]; __syncthreads(); }
    if (tid == 0) partial[blockIdx.x] = red[0];
}

// ---------------------------------------------------------------------------
// Kernel 4: deterministic final reduction + combine.
// out = sum(edge) - 0.5*e^2*sum(pair)
// ---------------------------------------------------------------------------
__global__ __launch_bounds__(256)
void LSM_final_kernel(const float* __restrict__ pairP,
                      const float* __restrict__ edgeP,
                      float* __restrict__ out)
{
    __shared__ float red[256];
    const int tid = threadIdx.x;
    float sp = 0.0f, se = 0.0f;
    for (int i = tid; i < NPB; i += 256) sp += pairP[i];
    for (int i = tid; i < NEB; i += 256) se += edgeP[i];
    red[tid] = sp; __syncthreads();
    for (int s = 128; s; s >>= 1) { if (tid < s) red[tid] += red[tid + s]; __syncthreads(); }
    const float SP = red[0];
    __syncthreads();
    red[tid] = se; __syncthreads();
    for (int s = 128; s; s >>= 1) { if (tid < s) red[tid] += red[tid + s]; __syncthreads(); }
    if (tid == 0) out[0] = red[0] - 0.5f * 7.3890560989306495f * SP;
}

extern "C" void kernel_launch(void* const* d_in, const int* in_sizes, int n_in,
                              void* d_out, int out_size, void* d_ws, size_t ws_size,
                              hipStream_t stream) {
    const float* beta = (const float*)d_in[0];
    const float* Z    = (const float*)d_in[1];
    const int*   sidx = (const int*)d_in[2];
    const int*   ei   = (const int*)d_in[3];
    const int*   ej   = (const int*)d_in[4];

    float* ws    = (float*)d_ws;
    float* na    = ws;                 // S_TOT
    float* nz    = ws + S_TOT;         // S_TOT
    float* bs    = ws + 2 * S_TOT;     // S_TOT
    float* pairP = ws + 3 * S_TOT;     // NPB
    float* edgeP = pairP + NPB;        // NEB   (total ~200 KB)

    LSM_norms_kernel<<<S_TOT / 8, 256, 0, stream>>>(beta, Z, sidx, na, nz, bs);
    dim3 grid(GRID1, GRID1);
    LSM_pair_kernel<<<grid, 256, 0, stream>>>(Z, sidx, na, nz, bs, pairP);
    LSM_edge_kernel<<<NEB, 256, 0, stream>>>(beta, Z, ei, ej, edgeP);
    LSM_final_kernel<<<1, 256, 0, stream>>>(pairP, edgeP, (float*)d_out);
}